// GaussianTPVCrossAttention_81346680586523
// MI455X (gfx1250) — compile-verified
//
#include <hip/hip_runtime.h>
#include <hip/hip_bf16.h>
#include <math.h>

// ---- problem constants (from reference) ----
#define B_      2
#define NQ_     2048
#define C_      256
#define NH_     8
#define HD_     32
#define A_      4
#define P_      5
#define S_      20      // A*P
#define L_      3
#define SUMHW_  46400   // 200*200 + 200*16 + 200*16
#define NEG_INF_ (-1000000000.0f)
#define LN_EPS_ 1e-5f

typedef float v2f __attribute__((ext_vector_type(2)));
typedef float v8f __attribute__((ext_vector_type(8)));

// Only check for the builtin in the DEVICE pass; the host pass (x86) never
// has amdgcn builtins and must not trip this.
#if defined(__HIP_DEVICE_COMPILE__)
#if !__has_builtin(__builtin_amdgcn_wmma_f32_16x16x4_f32)
#error "gfx1250 f32 WMMA builtin __builtin_amdgcn_wmma_f32_16x16x4_f32 not available"
#endif
#endif

__device__ __forceinline__ v8f wmma_f32x4(v2f a, v2f b, v8f c) {
  // 8 args: (neg_a, A, neg_b, B, c_mod, C, reuse_a, reuse_b)
  return __builtin_amdgcn_wmma_f32_16x16x4_f32(false, a, false, b, (short)0, c, false, false);
}

// ---------------------------------------------------------------------------
// fp32 GEMM with bias:  Cm[M,NN] = A[M,KK] @ Bw[KK,NN] + bias[NN]
// Wave tile: 32(M) x (NSUB*16)(N); K-step 4 via V_WMMA_F32_16X16X4_F32.
// NSUB/NN/KK are compile-time -> branch-free inner loop with constant
// address strides (no per-iteration 64-bit multiply for the B row pointer).
// 4 waves/block, each wave owns 2 consecutive M-tiles => block covers 128 rows
// (M is a multiple of 128 for all shapes used here: 92800, 4096).
// Each B fragment feeds 2 WMMAs (both M-subtiles) -> halves B-load issue.
// A fragment (16x4 fp32): lanes 0-15 -> K=k,k+1 ; lanes 16-31 -> K=k+2,k+3.
// B fragment (4x16 fp32): same K split, N = lane&15.
// C/D (16x16 fp32): VGPR i -> row i (lanes 0-15) / row i+8 (lanes 16-31).
// ---------------------------------------------------------------------------
template <int NSUB, int NN, int KK>
__global__ __launch_bounds__(128)
void gemm_bias_wmma(const float* __restrict__ A, const float* __restrict__ Bw,
                    const float* __restrict__ bias, float* __restrict__ Cm,
                    int M, int n0base) {
  (void)M;
  const int lane = threadIdx.x & 31;
  const int wave = threadIdx.x >> 5;
  const int mt   = (blockIdx.y * 4 + wave) * 2;       // first of 2 M-tiles
  const int n0   = n0base + blockIdx.x * (NSUB * 16);

  const int mrow0 = mt * 16 + (lane & 15);
  const int klo   = (lane >> 4) << 1;                 // 0 or 2
  const int nlane = lane & 15;

  v8f acc0[NSUB];
  v8f acc1[NSUB];
#pragma unroll
  for (int j = 0; j < NSUB; ++j) { acc0[j] = {}; acc1[j] = {}; }

  const float* __restrict__ arow0 = A + (size_t)mrow0 * KK + klo;
  const float* __restrict__ arow1 = arow0 + (size_t)16 * KK;
  const float* __restrict__ brow  = Bw + (size_t)klo * NN + n0 + nlane;
#pragma unroll 2
  for (int k = 0; k < KK; k += 4) {
    v2f a0 = *(const v2f*)(arow0 + k);                // 8B aligned (KK even)
    v2f a1 = *(const v2f*)(arow1 + k);
    const float* bp = brow + (size_t)k * NN;          // constant stride
#pragma unroll
    for (int j = 0; j < NSUB; ++j) {
      v2f b;
      b.x = bp[j * 16];
      b.y = bp[NN + j * 16];
      acc0[j] = wmma_f32x4(a0, b, acc0[j]);
      acc1[j] = wmma_f32x4(a1, b, acc1[j]);
    }
  }

  const int rbase0 = mt * 16 + ((lane >> 4) << 3);
  const int rbase1 = rbase0 + 16;
#pragma unroll
  for (int j = 0; j < NSUB; ++j) {
    const int col = n0 + j * 16 + nlane;
    const float bv = bias[col];
#pragma unroll
    for (int i = 0; i < 8; ++i) {
      Cm[(size_t)(rbase0 + i) * NN + col] = acc0[j][i] + bv;
      Cm[(size_t)(rbase1 + i) * NN + col] = acc1[j][i] + bv;
    }
  }
}

// ---------------------------------------------------------------------------
// Softmax (over S=20 with validity masking) + bilinear sampling.
// One wave per (b, q, head, level). Phase 1: lane = sample index s (<20):
// compute loc, validity, masked softmax via wave reductions. Phase 2: lane =
// channel d (HD=32 == wave32), broadcast per-sample scalars with __shfl and
// gather 4 bilinear corners from the projected value tensor (L2-resident).
// ---------------------------------------------------------------------------
__device__ __forceinline__ float gather_corner(const float* __restrict__ base,
                                               int xi, int yi, int W, int H) {
  bool inb = (xi >= 0) & (xi < W) & (yi >= 0) & (yi < H);
  int xc = min(max(xi, 0), W - 1);
  int yc = min(max(yi, 0), H - 1);
  float v = base[(size_t)(yc * W + xc) * (NH_ * HD_)];
  return inb ? v : 0.0f;
}

__global__ __launch_bounds__(256)
void sample_softmax_kernel(const float* __restrict__ val,      // (B,SUMHW,NH,HD)
                           const float* __restrict__ logits,   // L x (B*NQ, NH*S)
                           const float* __restrict__ offs,     // L x (B*NQ, NH*S*2)
                           const float* __restrict__ ref_pts,  // (L,B,NQ,A,2)
                           float* __restrict__ fused) {        // (B*NQ, L*C)
  const int lane = threadIdx.x & 31;
  const int wid  = blockIdx.x * 8 + (threadIdx.x >> 5);
  const int l  = wid % L_;
  const int hh = (wid / L_) % NH_;
  const int bq = wid / (L_ * NH_);
  if (bq >= B_ * NQ_) return;
  const int b = bq / NQ_;

  int Wl, start;
  if (l == 0)      { Wl = 200; start = 0; }
  else if (l == 1) { Wl = 16;  start = 40000; }
  else             { Wl = 16;  start = 43200; }
  const int Hl = 200;
  const float invW = 1.0f / (float)Wl;
  const float invH = 1.0f / (float)Hl;

  const float* logits_l = logits + (size_t)l * ((size_t)B_ * NQ_ * NH_ * S_);
  const float* offs_l   = offs   + (size_t)l * ((size_t)B_ * NQ_ * NH_ * S_ * 2);

  // ---- phase 1: lane == sample index s ----
  float lx = 0.f, ly = 0.f, logit = NEG_INF_;
  int validf = 0;
  if (lane < S_) {
    const int a    = lane / P_;
    const int col2 = (hh * S_ + lane) * 2;
    float ox = offs_l[(size_t)bq * (NH_ * S_ * 2) + col2];
    float oy = offs_l[(size_t)bq * (NH_ * S_ * 2) + col2 + 1];
    const float* rp = ref_pts + ((((size_t)l * B_ * NQ_ + bq) * A_) + a) * 2;
    lx = rp[0] + ox * invW;
    ly = rp[1] + oy * invH;
    validf = (lx >= 0.f) & (lx <= 1.f) & (ly >= 0.f) & (ly <= 1.f);
    logit = logits_l[(size_t)bq * (NH_ * S_) + hh * S_ + lane];
  }
  unsigned long long bal = __ballot(validf);
  if (bal == 0ull && lane == 0) validf = 1;        // all-invalid fallback

  float ml = validf ? logit : NEG_INF_;
  float mx = ml;
  for (int o = 16; o > 0; o >>= 1) mx = fmaxf(mx, __shfl_xor(mx, o, 32));
  float e = validf ? __expf(logit - mx) : 0.0f;
  float sum = e;
  for (int o = 16; o > 0; o >>= 1) sum += __shfl_xor(sum, o, 32);
  const float attn = e / sum;

  // ---- phase 2: lane == channel d ----
  const float* vbase =
      val + (((size_t)b * SUMHW_ + start) * NH_ + hh) * HD_ + lane;
  float acc = 0.f;
  for (int s = 0; s < S_; ++s) {
    float a_s = __shfl(attn, s, 32);
    float gx  = __shfl(lx, s, 32);
    float gy  = __shfl(ly, s, 32);
    gx = fminf(fmaxf(gx * 2.f - 1.f, -1.f + 1e-6f), 1.f - 1e-6f);
    gy = fminf(fmaxf(gy * 2.f - 1.f, -1.f + 1e-6f), 1.f - 1e-6f);
    float x = ((gx + 1.f) * (float)Wl - 1.f) * 0.5f;
    float y = ((gy + 1.f) * (float)Hl - 1.f) * 0.5f;
    float x0f = floorf(x), y0f = floorf(y);
    float fx = x - x0f, fy = y - y0f;
    int x0 = (int)x0f, y0 = (int)y0f;
    float w00 = (1.f - fx) * (1.f - fy);
    float w10 = fx * (1.f - fy);
    float w01 = (1.f - fx) * fy;
    float w11 = fx * fy;
    acc += a_s * (w00 * gather_corner(vbase, x0,     y0,     Wl, Hl) +
                  w10 * gather_corner(vbase, x0 + 1, y0,     Wl, Hl) +
                  w01 * gather_corner(vbase, x0,     y0 + 1, Wl, Hl) +
                  w11 * gather_corner(vbase, x0 + 1, y0 + 1, Wl, Hl));
  }
  fused[(size_t)bq * (L_ * C_) + l * C_ + hh * HD_ + lane] = acc;
}

// ---------------------------------------------------------------------------
// out = LayerNorm(query + pre) * g + b ; one 256-thread block per row (C=256)
// ---------------------------------------------------------------------------
__global__ __launch_bounds__(256)
void add_layernorm_kernel(const float* __restrict__ pre,
                          const float* __restrict__ query,
                          const float* __restrict__ g,
                          const float* __restrict__ bb,
                          float* __restrict__ out) {
  const int row = blockIdx.x;
  const int t   = threadIdx.x;
  float x = query[(size_t)row * C_ + t] + pre[(size_t)row * C_ + t];
  float s1 = x, s2 = x * x;
  for (int o = 16; o > 0; o >>= 1) {
    s1 += __shfl_xor(s1, o, 32);
    s2 += __shfl_xor(s2, o, 32);
  }
  __shared__ float sh1[8], sh2[8];
  if ((t & 31) == 0) { sh1[t >> 5] = s1; sh2[t >> 5] = s2; }
  __syncthreads();
  float S1 = 0.f, S2 = 0.f;
#pragma unroll
  for (int i = 0; i < 8; ++i) { S1 += sh1[i]; S2 += sh2[i]; }
  const float m = S1 * (1.0f / C_);
  const float v = S2 * (1.0f / C_) - m * m;
  out[(size_t)row * C_ + t] = (x - m) * rsqrtf(v + LN_EPS_) * g[t] + bb[t];
}

// ---------------------------------------------------------------------------
extern "C" void kernel_launch(void* const* d_in, const int* in_sizes, int n_in,
                              void* d_out, int out_size, void* d_ws, size_t ws_size,
                              hipStream_t stream) {
  (void)in_sizes; (void)n_in; (void)out_size; (void)ws_size;
  const float* query   = (const float*)d_in[0];   // (B,NQ,C)
  const float* value   = (const float*)d_in[1];   // (B,SUMHW,C)
  const float* ref_pts = (const float*)d_in[2];   // (L,B,NQ,A,2)
  const float* value_w = (const float*)d_in[3];   // (C,C)
  const float* value_b = (const float*)d_in[4];   // (C,)
  const float* att_w   = (const float*)d_in[5];   // (L,C,NH*S)
  const float* att_b   = (const float*)d_in[6];   // (L,NH*S)
  const float* off_w   = (const float*)d_in[7];   // (L,C,NH*S*2)
  const float* off_b   = (const float*)d_in[8];   // (L,NH*S*2)
  const float* out_w   = (const float*)d_in[9];   // (L*C,C)
  const float* out_b   = (const float*)d_in[10];  // (C,)
  const float* ln_g    = (const float*)d_in[11];
  const float* ln_b    = (const float*)d_in[12];

  // workspace layout (floats)
  float* ws = (float*)d_ws;
  const size_t VAL_N   = (size_t)B_ * SUMHW_ * C_;            // 23,756,800
  const size_t LOGIT_N = (size_t)L_ * B_ * NQ_ * NH_ * S_;    //  1,966,080
  const size_t OFF_N   = (size_t)L_ * B_ * NQ_ * NH_ * S_ * 2;//  3,932,160
  const size_t FUSED_N = (size_t)B_ * NQ_ * L_ * C_;          //  3,145,728
  float* valb  = ws;
  float* logit = valb + VAL_N;
  float* offv  = logit + LOGIT_N;
  float* fused = offv + OFF_N;
  float* pre   = fused + FUSED_N;

  const dim3 blk(128);

  // 1) value projection: (92800,256) @ (256,256) + b   [M/128 = 725]
  hipLaunchKernelGGL((gemm_bias_wmma<4, 256, 256>), dim3(4, 725), blk, 0, stream,
                     value, value_w, value_b, valb, B_ * SUMHW_, 0);

  // 2) per-level attention logits (N=160) and offsets (N=320), M=4096 [M/128=32]
  for (int l = 0; l < L_; ++l) {
    const int M = B_ * NQ_;
    {
      const int N = NH_ * S_;                       // 160 = 2*64 + 32
      const float* Bw = att_w + (size_t)l * C_ * N;
      const float* bs = att_b + (size_t)l * N;
      float* Cmat = logit + (size_t)l * M * N;
      hipLaunchKernelGGL((gemm_bias_wmma<4, 160, 256>), dim3(2, 32), blk, 0, stream,
                         query, Bw, bs, Cmat, M, 0);
      hipLaunchKernelGGL((gemm_bias_wmma<2, 160, 256>), dim3(1, 32), blk, 0, stream,
                         query, Bw, bs, Cmat, M, 128);
    }
    {
      const int N = NH_ * S_ * 2;                   // 320 = 5*64
      const float* Bw = off_w + (size_t)l * C_ * N;
      const float* bs = off_b + (size_t)l * N;
      float* Cmat = offv + (size_t)l * M * N;
      hipLaunchKernelGGL((gemm_bias_wmma<4, 320, 256>), dim3(5, 32), blk, 0, stream,
                         query, Bw, bs, Cmat, M, 0);
    }
  }

  // 3) softmax + bilinear sampling -> fused (B*NQ, 768)
  {
    int waves = B_ * NQ_ * NH_ * L_;                // 98304, divisible by 8
    hipLaunchKernelGGL(sample_softmax_kernel, dim3(waves / 8), dim3(256), 0, stream,
                       valb, logit, offv, ref_pts, fused);
  }

  // 4) output projection: (4096,768) @ (768,256) + b
  hipLaunchKernelGGL((gemm_bias_wmma<4, 256, 768>), dim3(4, 32), blk, 0, stream,
                     fused, out_w, out_b, pre, B_ * NQ_, 0);

  // 5) residual + layernorm -> d_out
  hipLaunchKernelGGL(add_layernorm_kernel, dim3(B_ * NQ_), dim3(256), 0, stream,
                     pre, query, ln_g, ln_b, (float*)d_out);
}